// SinkhornDistance_75213467288081
// MI455X (gfx1250) — compile-verified
//
#include <hip/hip_runtime.h>

// Sinkhorn distance, MI455X (gfx1250, wave32).
// x,y: (8, 2048, 64) f32.  Outputs concat: dis(8*2048), cost(8), pi(8*2048*2048), C(8*2048*2048).

#define NB 8
#define P  2048
#define D  64
#define EPS_ 0.1f
#define INV_EPS 10.0f
#define THRESH_ 0.1f
#define NITER 100
#define SEGS 16
#define SEGROWS (P / SEGS) // 128
// log(1/2048 + 1e-8)
#define LOG_MU -7.6245985f

typedef float v2f __attribute__((ext_vector_type(2)));
typedef float v8f __attribute__((ext_vector_type(8)));

// ---------------- init: zero accumulators / state (runs every call) ----------------
__global__ void init_kernel(float* __restrict__ u, float* __restrict__ v,
                            float* __restrict__ err, int* __restrict__ done,
                            float* __restrict__ dis, float* __restrict__ cost) {
  int idx = blockIdx.x * 256 + threadIdx.x;
  if (idx < NB * P) { u[idx] = 0.f; v[idx] = 0.f; dis[idx] = 0.f; }
  if (idx < 128)    { err[idx] = 0.f; done[idx] = 0; }
  if (idx < NB)     { cost[idx] = 0.f; }
}

// ---------------- row squared-norm means: x2[b,i] = mean_k x^2 ----------------
__global__ void norms_kernel(const float* __restrict__ x, const float* __restrict__ y,
                             float* __restrict__ x2, float* __restrict__ y2) {
  int lane = threadIdx.x & 31, wid = threadIdx.x >> 5;
  int r = blockIdx.x * 8 + wid; // 0..32767 : first 16384 rows of x, then y
  const float* src; float* dst; int row;
  if (r < NB * P) { src = x; dst = x2; row = r; }
  else            { src = y; dst = y2; row = r - NB * P; }
  const float* p = src + (size_t)row * D;
  float a0 = p[lane], a1 = p[lane + 32];
  float s = a0 * a0 + a1 * a1;
  #pragma unroll
  for (int off = 16; off; off >>= 1) s += __shfl_xor(s, off, 32);
  if (lane == 0) dst[row] = s * (1.0f / D);
}

// ---------------- cost matrix via f32 WMMA + async LDS staging ----------------
// Block (256 thr / 8 waves) computes a 128x128 C tile.
// x/y 128x64 panels staged into LDS with GLOBAL_LOAD_ASYNC_TO_LDS_B128 (ASYNCcnt),
// 16-byte chunks XOR-swizzled (chunk ^= row&15) so the strided ds_load_b64 operand
// fetches are bank-conflict-free without padding. Each wave computes a 16x128 strip
// (8 accumulator tiles, A operand reused 8x per K-step).
__global__ void __launch_bounds__(256) gemm_kernel(
    const float* __restrict__ x, const float* __restrict__ y,
    const float* __restrict__ x2, const float* __restrict__ y2,
    float* __restrict__ C) {
  __shared__ float lx[128 * 64];
  __shared__ float ly[128 * 64];

  int blk = blockIdx.x;            // 0..2047 : 8 b * 16 I * 16 J
  int b   = blk >> 8;
  int rem = blk & 255;
  int I0  = (rem >> 4) << 7;
  int J0  = (rem & 15) << 7;

  const float* xb = x + ((size_t)b * P + I0) * D;
  const float* yb = y + ((size_t)b * P + J0) * D;
  unsigned lxa = (unsigned)(uintptr_t)&lx[0];   // low 32 bits of flat ptr == LDS byte addr
  unsigned lya = (unsigned)(uintptr_t)&ly[0];

  int tid = threadIdx.x;
  #pragma unroll
  for (int q = 0; q < 8; q++) {
    int e     = q * 256 + tid;               // 0..2047 16B-transfers per panel
    int row   = e >> 4;
    int chunk = e & 15;
    int chs   = chunk ^ (row & 15);          // xor-swizzle 16B chunks
    unsigned goff = (unsigned)((row * 64 + chunk * 4) * 4);
    unsigned lofx = lxa + (unsigned)(row * 256 + chs * 16);
    unsigned lofy = lya + (unsigned)(row * 256 + chs * 16);
    asm volatile("global_load_async_to_lds_b128 %0, %1, %2"
                 :: "v"(lofx), "v"(goff), "s"(xb) : "memory");
    asm volatile("global_load_async_to_lds_b128 %0, %1, %2"
                 :: "v"(lofy), "v"(goff), "s"(yb) : "memory");
  }
  asm volatile("s_wait_asynccnt 0x0" ::: "memory");
  __syncthreads();

  int lane = tid & 31, w = tid >> 5;
  int half = lane >> 4, l15 = lane & 15;

  v8f acc[8];
  #pragma unroll
  for (int jt = 0; jt < 8; jt++) acc[jt] = (v8f){};

  // ISA 7.12.2 f32 A(16x4)/B(4x16) layout: lanes 0-15 hold K={0,1}, lanes 16-31 K={2,3};
  // row/col = lane&15. Per K-step each lane reads an 8B K-pair from LDS.
  #pragma unroll
  for (int k0 = 0; k0 < D; k0 += 4) {
    int boff = ((k0 >> 2) ^ l15) * 4 + 2 * half;   // swizzled float offset in row
    v2f a = *(const v2f*)&lx[(w * 16 + l15) * 64 + boff];
    #pragma unroll
    for (int jt = 0; jt < 8; jt++) {
      v2f bb = *(const v2f*)&ly[(jt * 16 + l15) * 64 + boff];
      acc[jt] = __builtin_amdgcn_wmma_f32_16x16x4_f32(false, a, false, bb,
                                                      (short)0, acc[jt], false, false);
    }
  }

  // C/D layout: N = lane&15; VGPR r holds M = r + 8*(lane>>4).
  const float scale = 2.0f / D;
  int i0g = I0 + w * 16;
  const float* x2b = x2 + b * P;
  const float* y2b = y2 + b * P;
  float x2v[8];
  #pragma unroll
  for (int r2 = 0; r2 < 8; r2++) x2v[r2] = x2b[i0g + r2 + 8 * half];

  #pragma unroll
  for (int jt = 0; jt < 8; jt++) {
    int j = J0 + jt * 16 + l15;
    float y2j = y2b[j];
    float* Cb = C + ((size_t)b * P + i0g) * P + j;
    #pragma unroll
    for (int r2 = 0; r2 < 8; r2++) {
      int i = r2 + 8 * half;
      Cb[(size_t)i * P] = x2v[r2] + y2j - scale * acc[jt][r2];
    }
  }
}

// ---------------- row pass: u_i <- eps*(log_mu - lse_j((u_i+v_j-C_ij)/eps)) + u_i ----------------
// One wave per row; full 8KB row cached in 64 VGPRs -> single L2 read, one exp/element.
__global__ void __launch_bounds__(256) row_kernel(
    const float* __restrict__ C, float* __restrict__ u, const float* __restrict__ v,
    float* __restrict__ err, const int* __restrict__ done, int t) {
  if (done[t]) return;
  int lane = threadIdx.x & 31, wid = threadIdx.x >> 5;
  int r = blockIdx.x * 8 + wid;    // b*P + i
  int b = r >> 11;
  const float* Crow = C + (size_t)r * P;
  const float* vb   = v + b * P;
  float ui = u[r];

  float vals[64];
  #pragma unroll
  for (int k = 0; k < 64; k++) {
    int j = lane + (k << 5);
    vals[k] = (ui + vb[j] - Crow[j]) * INV_EPS;
  }
  float m = vals[0];
  #pragma unroll
  for (int k = 1; k < 64; k++) m = fmaxf(m, vals[k]);
  #pragma unroll
  for (int off = 16; off; off >>= 1) m = fmaxf(m, __shfl_xor(m, off, 32));
  float s = 0.f;
  #pragma unroll
  for (int k = 0; k < 64; k++) s += __expf(vals[k] - m);
  #pragma unroll
  for (int off = 16; off; off >>= 1) s += __shfl_xor(s, off, 32);

  if (lane == 0) {
    float lse = m + __logf(s);
    float un  = EPS_ * (LOG_MU - lse) + ui;
    u[r] = un;
    atomicAdd(err + t, fabsf(un - ui));
  }
}

// ---------------- col pass stage 1: partial (max, sumexp) over 128-row segments ----------------
__global__ void __launch_bounds__(256) col1_kernel(
    const float* __restrict__ C, const float* __restrict__ u, const float* __restrict__ v,
    float* __restrict__ pm, float* __restrict__ ps, const int* __restrict__ done, int t) {
  if (done[t]) return;
  int blk = blockIdx.x;            // 0..1023
  int seg = blk & 15;
  int bc  = blk >> 4;              // 0..63
  int b   = bc >> 3;
  int jc  = bc & 7;
  int j   = jc * 256 + threadIdx.x;
  float vj = v[b * P + j];
  const float* Cb = C + (size_t)b * P * P;
  const float* ub = u + b * P;
  int i0 = seg * SEGROWS;

  float m = -1e30f;
  for (int ii = 0; ii < SEGROWS; ii++) {
    int i = i0 + ii;
    float tt = (ub[i] + vj - Cb[(size_t)i * P + j]) * INV_EPS;
    m = fmaxf(m, tt);
  }
  float s = 0.f;
  for (int ii = 0; ii < SEGROWS; ii++) {
    int i = i0 + ii;
    float tt = (ub[i] + vj - Cb[(size_t)i * P + j]) * INV_EPS;
    s += __expf(tt - m);
  }
  size_t pidx = (size_t)seg * (NB * P) + b * P + j;
  pm[pidx] = m; ps[pidx] = s;
}

// ---------------- col pass stage 2: combine segments, update v ----------------
__global__ void __launch_bounds__(256) col2_kernel(
    float* __restrict__ v, const float* __restrict__ pm, const float* __restrict__ ps,
    const int* __restrict__ done, int t) {
  if (done[t]) return;
  int j = blockIdx.x * 256 + threadIdx.x;   // b*P + jj
  float M = -1e30f;
  #pragma unroll
  for (int seg = 0; seg < SEGS; seg++) M = fmaxf(M, pm[(size_t)seg * (NB * P) + j]);
  float S = 0.f;
  #pragma unroll
  for (int seg = 0; seg < SEGS; seg++) {
    size_t pidx = (size_t)seg * (NB * P) + j;
    S += ps[pidx] * __expf(pm[pidx] - M);
  }
  float lse = M + __logf(S);
  float vo  = v[j];
  v[j] = EPS_ * (LOG_MU - lse) + vo;        // log_nu == log_mu here
}

// ---------------- convergence flag chain ----------------
__global__ void flag_kernel(const float* __restrict__ err, int* __restrict__ done, int t) {
  if (threadIdx.x == 0)
    done[t + 1] = done[t] | ((err[t] * (1.0f / NB) < THRESH_) ? 1 : 0);
}

// ---------------- final: pi = exp(M(u,v)); dis = sum_i pi*C; cost = sum_ij pi*C ----------------
__global__ void __launch_bounds__(256) final_kernel(
    const float* __restrict__ C, const float* __restrict__ u, const float* __restrict__ v,
    float* __restrict__ pi, float* __restrict__ dis, float* __restrict__ cost) {
  __shared__ float red[256];
  int blk = blockIdx.x;            // 1024 blocks: 8b x 8jc x 16seg
  int seg = blk & 15;
  int bc  = blk >> 4;
  int b   = bc >> 3;
  int jc  = bc & 7;
  int j   = jc * 256 + threadIdx.x;
  float vj = v[b * P + j];
  const float* ub = u + b * P;
  const float* Cb = C  + (size_t)b * P * P;
  float*      pib = pi + (size_t)b * P * P;

  float acc = 0.f;
  for (int ii = 0; ii < SEGROWS; ii++) {
    int i = seg * SEGROWS + ii;
    size_t idx = (size_t)i * P + j;
    float c = Cb[idx];
    float p = __expf((ub[i] + vj - c) * INV_EPS);
    pib[idx] = p;
    acc += p * c;
  }
  atomicAdd(&dis[b * P + j], acc);

  red[threadIdx.x] = acc;
  __syncthreads();
  for (int off = 128; off; off >>= 1) {
    if (threadIdx.x < off) red[threadIdx.x] += red[threadIdx.x + off];
    __syncthreads();
  }
  if (threadIdx.x == 0) atomicAdd(&cost[b], red[0]);
}

// ---------------- launch ----------------
extern "C" void kernel_launch(void* const* d_in, const int* in_sizes, int n_in,
                              void* d_out, int out_size, void* d_ws, size_t ws_size,
                              hipStream_t stream) {
  const float* x = (const float*)d_in[0];
  const float* y = (const float*)d_in[1];

  float* out  = (float*)d_out;
  float* dis  = out;                         // 8*2048
  float* cost = out + NB * P;                // 8
  float* pi   = out + NB * P + NB;           // 8*2048*2048
  float* C    = pi + (size_t)NB * P * P;     // 8*2048*2048

  float* ws = (float*)d_ws;
  float* u   = ws;                           // 16384
  float* v   = ws + 16384;                   // 16384
  float* x2  = ws + 32768;                   // 16384
  float* y2  = ws + 49152;                   // 16384
  float* err = ws + 65536;                   // 128
  int*   done = (int*)(ws + 65664);          // 128
  float* pm  = ws + 65792;                   // 16*16384
  float* ps  = pm + (size_t)SEGS * NB * P;   // 16*16384

  init_kernel <<<64,   256, 0, stream>>>(u, v, err, done, dis, cost);
  norms_kernel<<<4096, 256, 0, stream>>>(x, y, x2, y2);
  gemm_kernel <<<2048, 256, 0, stream>>>(x, y, x2, y2, C);

  for (int t = 0; t < NITER; t++) {
    row_kernel <<<2048, 256, 0, stream>>>(C, u, v, err, done, t);
    col1_kernel<<<1024, 256, 0, stream>>>(C, u, v, pm, ps, done, t);
    col2_kernel<<<64,   256, 0, stream>>>(v, pm, ps, done, t);
    flag_kernel<<<1,    32,  0, stream>>>(err, done, t);
  }

  final_kernel<<<1024, 256, 0, stream>>>(C, u, v, pi, dis, cost);
}